// MaSIF_site2_35570919146273
// MI455X (gfx1250) — compile-verified
//
#include <hip/hip_runtime.h>
#include <math.h>

// Problem constants (match reference)
#define Bc 8
#define Sc 500
#define Vc 100
#define Fc 5
#define Kc 12
#define FK 60           // F*K
#define EPSc 1e-5f
#define TWO_PI_F 6.283185307179586f
#define INV_TWO_PI_F 0.15915494309189535f

typedef __attribute__((ext_vector_type(2))) float v2f;
typedef __attribute__((ext_vector_type(8))) float v8f;

// V_WMMA_F32_16X16X4_F32 : D(16x16,f32) = A(16x4,f32) * B(4x16,f32) + C
__device__ __forceinline__ v8f wmma4(v2f a, v2f b, v8f c) {
    return __builtin_amdgcn_wmma_f32_16x16x4_f32(false, a, false, b, (short)0, c,
                                                 false, false);
}

// ---------------------------------------------------------------------------
// Pass 1: denominators D[b,f,v,k] = sum_s  g_rho * g_theta * mask
// (reference normalizes over axis=2 == S of the (B,F,S,V,K) tensor)
// grid = B*V blocks, 64 threads. Threads 0..59 own one (f,k) pair.
// ---------------------------------------------------------------------------
__global__ void masif_pass1_denom(const float* __restrict__ rho,
                                  const float* __restrict__ theta,
                                  const float* __restrict__ mask,
                                  const float* __restrict__ mu_rho,
                                  const float* __restrict__ sigma_rho,
                                  const float* __restrict__ mu_theta,
                                  const float* __restrict__ sigma_theta,
                                  float* __restrict__ D) {
    __shared__ float s_rho[Sc];
    __shared__ float s_th[Sc];
    __shared__ float s_m[Sc];

    const int b = blockIdx.x / Vc;
    const int v = blockIdx.x % Vc;
    const int tid = threadIdx.x;

    // stage rho/theta/mask for this (b, :, v) column (stride V in memory)
    for (int s = tid; s < Sc; s += blockDim.x) {
        const int idx = (b * Sc + s) * Vc + v;
        s_rho[s] = rho[idx];
        float th = theta[idx];
        th = th - TWO_PI_F * floorf(th * INV_TWO_PI_F);  // jnp.mod(th, 2pi), r=0
        s_th[s] = th;
        s_m[s] = mask[idx];
    }
    __syncthreads();

    if (tid < FK) {
        const int f = tid / Kc;
        const int k = tid % Kc;
        const float mr = mu_rho[f * Kc + k];
        const float sr = sigma_rho[f * Kc + k];
        const float mt = mu_theta[f * Kc + k];
        const float st = sigma_theta[f * Kc + k];
        const float inv_r = 1.0f / (sr * sr + EPSc);
        const float inv_t = 1.0f / (st * st + EPSc);

        float acc = 0.0f;
        #pragma unroll 4
        for (int s = 0; s < Sc; ++s) {
            const float dr = s_rho[s] - mr;
            const float dt = s_th[s] - mt;
            acc += __expf(-(dr * dr) * inv_r) * __expf(-(dt * dt) * inv_t) * s_m[s];
        }
        D[((b * Fc + f) * Vc + v) * Kc + k] = acc;
    }
}

// ---------------------------------------------------------------------------
// Pass 2: desc[b,s,f,k] = sum_v ( g / (D[b,f,v,k]+eps) ) * feat[b,s,v,f]
// grid = B*S blocks, 64 threads. Threads 0..59 own one (f,k) pair.
// Output DESC laid out row-major (b*S+s) x 60, col index = f*12+k.
// ---------------------------------------------------------------------------
__global__ void masif_pass2_desc(const float* __restrict__ feat,
                                 const float* __restrict__ rho,
                                 const float* __restrict__ theta,
                                 const float* __restrict__ mask,
                                 const float* __restrict__ mu_rho,
                                 const float* __restrict__ sigma_rho,
                                 const float* __restrict__ mu_theta,
                                 const float* __restrict__ sigma_theta,
                                 const float* __restrict__ D,
                                 float* __restrict__ DESC) {
    __shared__ float l_rho[Vc];
    __shared__ float l_th[Vc];
    __shared__ float l_m[Vc];
    __shared__ float l_feat[Vc * Fc];

    const int bs = blockIdx.x;          // b*S + s
    const int b = bs / Sc;
    const int tid = threadIdx.x;

    const int base = bs * Vc;           // rho/theta/mask base
    for (int v = tid; v < Vc; v += blockDim.x) {
        l_rho[v] = rho[base + v];
        float th = theta[base + v];
        th = th - TWO_PI_F * floorf(th * INV_TWO_PI_F);
        l_th[v] = th;
        l_m[v] = mask[base + v];
    }
    const int fbase = bs * Vc * Fc;     // feat base, layout [v][f]
    for (int i = tid; i < Vc * Fc; i += blockDim.x) l_feat[i] = feat[fbase + i];
    __syncthreads();

    if (tid < FK) {
        const int f = tid / Kc;
        const int k = tid % Kc;
        const float mr = mu_rho[f * Kc + k];
        const float sr = sigma_rho[f * Kc + k];
        const float mt = mu_theta[f * Kc + k];
        const float st = sigma_theta[f * Kc + k];
        const float inv_r = 1.0f / (sr * sr + EPSc);
        const float inv_t = 1.0f / (st * st + EPSc);
        const long Dbase = (long)((b * Fc + f) * Vc) * Kc + k;

        float acc = 0.0f;
        #pragma unroll 4
        for (int v = 0; v < Vc; ++v) {
            const float dr = l_rho[v] - mr;
            const float dt = l_th[v] - mt;
            float g = __expf(-(dr * dr) * inv_r) * __expf(-(dt * dt) * inv_t) * l_m[v];
            g = g / (D[Dbase + (long)v * Kc] + EPSc);
            acc += g * l_feat[v * Fc + f];
        }
        DESC[(long)bs * FK + tid] = acc;   // tid == f*12 + k
    }
}

// ---------------------------------------------------------------------------
// Head: per-f conv (12x12) + relu, then 60->12 relu, 12->5 relu, 5->3 softmax.
// One wave (32 lanes) per 16-row tile of the 4000-row batch.
//
// WMMA f32 16x16x4 operand layout (wave32):
//   A: lane L holds row M = L&15, K pair k0 = (L>>4)*2  -> {A[M][k0], A[M][k0+1]}
//   B: lane L holds col N = L&15, K pair k0 = (L>>4)*2  -> {B[k0][N], B[k0+1][N]}
//   C/D: lane L col N = L&15; vgpr i -> row i + (L>>4)*8
//
// All weights are staged in LDS zero-padded and PAIR-INTERLEAVED:
//   sW[(pair*16 + n)*2 + w] = B[2*pair + w][n]  (n >= valid_cols -> 0)
// so each B operand is a single unconditional 8-byte ds load — no exec-mask
// branches anywhere in the WMMA stream (EXEC stays all-ones, per ISA rules).
// ---------------------------------------------------------------------------
__global__ void masif_head_wmma(const float* __restrict__ DESC,
                                const float* __restrict__ Wc,   // (F,12,12)
                                const float* __restrict__ bc,   // (F,12)
                                const float* __restrict__ W2,   // (60,12)
                                const float* __restrict__ b2,   // (12)
                                const float* __restrict__ W3,   // (12,5)
                                const float* __restrict__ b3,   // (5)
                                const float* __restrict__ W4,   // (5,3)
                                const float* __restrict__ b4,   // (3)
                                float* __restrict__ out) {      // (4000,3)
    __shared__ __align__(16) float sWc[Fc * 6 * 16 * 2];  // 960: f, pair, n, w
    __shared__ __align__(16) float sW2[30 * 16 * 2];      // 960: pair, n, w
    __shared__ __align__(16) float sW3[6 * 16 * 2];       // 192: pair, n, w
    __shared__ __align__(16) float sBc[Fc * 16];          // padded conv bias
    __shared__ __align__(16) float sB2[16];
    __shared__ __align__(16) float sB3[16];
    __shared__ __align__(16) float gd[16 * FK];           // relu(conv), 16 x 60
    __shared__ __align__(16) float h1[16 * 16];           // layer-2 out (cols<12)
    __shared__ __align__(16) float h2[16 * 8];            // layer-3 out (cols<5)

    const int tile = blockIdx.x;            // 250 tiles of 16 rows
    const int lane = threadIdx.x;           // 0..31, wave32
    const int N = lane & 15;                // B/C column this lane owns
    const int half = lane >> 4;
    const int k0 = half * 2;                // K pair owned by this half-wave
    const int M = lane & 15;                // A row this lane owns
    const int row0 = tile * 16;

    // ---- stage zero-padded, pair-interleaved weights into LDS --------------
    for (int i = lane; i < Fc * 192; i += 32) {           // sWc
        const int f = i / 192, j = i % 192;
        const int p = j >> 5, n = (j & 31) >> 1, w = j & 1;
        sWc[i] = (n < Kc) ? Wc[f * (Kc * Kc) + (2 * p + w) * Kc + n] : 0.0f;
    }
    for (int i = lane; i < 960; i += 32) {                // sW2
        const int p = i >> 5, n = (i & 31) >> 1, w = i & 1;
        sW2[i] = (n < Kc) ? W2[(2 * p + w) * Kc + n] : 0.0f;
    }
    for (int i = lane; i < 192; i += 32) {                // sW3
        const int p = i >> 5, n = (i & 31) >> 1, w = i & 1;
        sW3[i] = (n < Fc) ? W3[(2 * p + w) * Fc + n] : 0.0f;
    }
    for (int i = lane; i < Fc * 16; i += 32)
        sBc[i] = ((i & 15) < Kc) ? bc[(i >> 4) * Kc + (i & 15)] : 0.0f;
    if (lane < 16) sB2[lane] = (lane < Kc) ? b2[lane] : 0.0f;
    if (lane < 16) sB3[lane] = (lane < Fc) ? b3[lane] : 0.0f;
    __syncthreads();

    // ---- per-f conv: desc(16x12) @ Wc[f](12x12), bias, relu -> gd ----------
    for (int f = 0; f < Fc; ++f) {
        v8f c = {};
        #pragma unroll
        for (int ch = 0; ch < 3; ++ch) {
            const v2f a =
                *(const v2f*)&DESC[(long)(row0 + M) * FK + f * Kc + ch * 4 + k0];
            const v2f b = *(const v2f*)&sWc[((f * 6 + ch * 2 + half) * 16 + N) * 2];
            c = wmma4(a, b, c);
        }
        const float bias = sBc[f * 16 + N];
        #pragma unroll
        for (int i = 0; i < 8; ++i) {
            const int m = i + half * 8;
            if (N < Kc) gd[m * FK + f * Kc + N] = fmaxf(c[i] + bias, 0.0f);
        }
    }
    __syncthreads();

    // ---- layer 2: gd(16x60) @ W2(60x12) + b2, relu -------------------------
    v8f c2 = {};
    #pragma unroll
    for (int ch = 0; ch < 15; ++ch) {
        const v2f a = *(const v2f*)&gd[M * FK + ch * 4 + k0];
        const v2f b = *(const v2f*)&sW2[((ch * 2 + half) * 16 + N) * 2];
        c2 = wmma4(a, b, c2);
    }
    {
        const float bias = sB2[N];
        #pragma unroll
        for (int i = 0; i < 8; ++i) {
            const int m = i + half * 8;
            h1[m * 16 + N] = (N < Kc) ? fmaxf(c2[i] + bias, 0.0f) : 0.0f;
        }
    }
    __syncthreads();

    // ---- layer 3: h1(16x12) @ W3(12x5) + b3, relu --------------------------
    v8f c3 = {};
    #pragma unroll
    for (int ch = 0; ch < 3; ++ch) {
        const v2f a = *(const v2f*)&h1[M * 16 + ch * 4 + k0];
        const v2f b = *(const v2f*)&sW3[((ch * 2 + half) * 16 + N) * 2];
        c3 = wmma4(a, b, c3);
    }
    {
        const float bias = sB3[N];
        #pragma unroll
        for (int i = 0; i < 8; ++i) {
            const int m = i + half * 8;
            if (N < Fc) h2[m * 8 + N] = fmaxf(c3[i] + bias, 0.0f);
        }
    }
    __syncthreads();

    // ---- layer 4 + softmax: (16x5) @ (5x3), 3 outputs per row --------------
    if (lane < 16) {
        float z[3];
        #pragma unroll
        for (int j = 0; j < 3; ++j) {
            float acc = b4[j];
            #pragma unroll
            for (int q = 0; q < Fc; ++q) acc += h2[lane * 8 + q] * W4[q * 3 + j];
            z[j] = acc;
        }
        float mx = fmaxf(z[0], fmaxf(z[1], z[2]));
        float e0 = __expf(z[0] - mx), e1 = __expf(z[1] - mx), e2 = __expf(z[2] - mx);
        const float inv = 1.0f / (e0 + e1 + e2);
        const long obase = (long)(row0 + lane) * 3;
        out[obase + 0] = e0 * inv;
        out[obase + 1] = e1 * inv;
        out[obase + 2] = e2 * inv;
    }
}

// ---------------------------------------------------------------------------
extern "C" void kernel_launch(void* const* d_in, const int* in_sizes, int n_in,
                              void* d_out, int out_size, void* d_ws, size_t ws_size,
                              hipStream_t stream) {
    const float* input_feat  = (const float*)d_in[0];   // (B,S,V,F)
    const float* rho_coords  = (const float*)d_in[1];   // (B,S,V,1)
    const float* theta_coords= (const float*)d_in[2];   // (B,S,V,1)
    const float* mask        = (const float*)d_in[3];   // (B,S,V,1)
    const float* mu_rho      = (const float*)d_in[4];   // (F,K)
    const float* sigma_rho   = (const float*)d_in[5];   // (F,K)
    const float* mu_theta    = (const float*)d_in[6];   // (F,K)
    const float* sigma_theta = (const float*)d_in[7];   // (F,K)
    const float* W_conv      = (const float*)d_in[8];   // (F,K,K)
    const float* b_conv      = (const float*)d_in[9];   // (F,K)
    const float* W2          = (const float*)d_in[10];  // (F*K,K)
    const float* b2          = (const float*)d_in[11];  // (K)
    const float* W3          = (const float*)d_in[12];  // (K,F)
    const float* b3          = (const float*)d_in[13];  // (F)
    const float* W4          = (const float*)d_in[14];  // (F,3)
    const float* b4          = (const float*)d_in[15];  // (3)
    float* out = (float*)d_out;                         // (B,S,3)

    // workspace: D = B*F*V*K floats (48000), DESC = B*S*60 floats (240000)
    float* D    = (float*)d_ws;
    float* DESC = D + (Bc * Fc * Vc * Kc);

    masif_pass1_denom<<<Bc * Vc, 64, 0, stream>>>(
        rho_coords, theta_coords, mask, mu_rho, sigma_rho, mu_theta, sigma_theta, D);

    masif_pass2_desc<<<Bc * Sc, 64, 0, stream>>>(
        input_feat, rho_coords, theta_coords, mask,
        mu_rho, sigma_rho, mu_theta, sigma_theta, D, DESC);

    masif_head_wmma<<<(Bc * Sc) / 16, 32, 0, stream>>>(
        DESC, W_conv, b_conv, W2, b2, W3, b3, W4, b4, out);
}